// GroupBilinearPoolingNet_78512002171089
// MI455X (gfx1250) — compile-verified
//
#include <hip/hip_runtime.h>

typedef float v2f __attribute__((ext_vector_type(2)));
typedef float v8f __attribute__((ext_vector_type(8)));

#define EPS_SQ 1e-12f
#define REC 36            // padded per-position record stride in LDS slab (bank-spread)

// Workspace layout (floats): Gram accumulators per level
#define G1_OFF 0          // [8][256][3*3]
#define G2_OFF 18432      // [8][128][4*4]
#define G3_OFF 34816      // [8][64][8*8]
#define G4_OFF 67584      // [8][32][16*16]
#define WS_FLOATS 133120

__device__ __forceinline__ void wait_ds() {
  asm volatile("s_wait_dscnt 0x0" ::: "memory");
}

__device__ __forceinline__ void atomAddF(float* p, float v) {
  __hip_atomic_fetch_add(p, v, __ATOMIC_RELAXED, __HIP_MEMORY_SCOPE_AGENT);
}

// ---- per-lane m-chain helpers (fully unrolled, register resident) ----
template<int C, int D>
__device__ __forceinline__ void proj(const float* __restrict__ P, const float* v, float* s) {
#pragma unroll
  for (int d = 0; d < D; ++d) {
    float a = 0.f;
#pragma unroll
    for (int c = 0; c < C; ++c) a = __builtin_fmaf(v[c], P[c * D + d], a);
    s[d] = a;
  }
}

template<int D>
__device__ __forceinline__ void cconv(const float* s1, const float* s2, float* o) {
#pragma unroll
  for (int d = 0; d < D; ++d) {
    float a = 0.f;
#pragma unroll
    for (int k = 0; k < D; ++k) a = __builtin_fmaf(s1[k], s2[(d - k + D) & (D - 1)], a);
    o[d] = a;
  }
}

template<int D>
__device__ __forceinline__ void ssn_norm(float* o) {
  float n2 = 0.f;
#pragma unroll
  for (int d = 0; d < D; ++d) {
    float t = o[d];
    float s = __builtin_sqrtf(__builtin_fabsf(t) + EPS_SQ);
    float y = t > 0.f ? s : (t < 0.f ? -s : 0.f);   // jnp.sign(0)==0
    o[d] = y;
    n2 = __builtin_fmaf(y, y, n2);
  }
  float inv = n2 > 0.f ? rsqrtf(n2) : 0.f;          // n >= 1e-6 whenever nonzero
#pragma unroll
  for (int d = 0; d < D; ++d) o[d] *= inv;
}

// ---- WMMA Gram machinery: lane-major -> WMMA A/B layout via per-wave LDS slab ----
template<int LEN>
__device__ __forceinline__ void stage(float* slab, int lane, const float* src) {
  float* rec = slab + lane * REC;
#pragma unroll
  for (int i = 0; i < LEN; ++i) rec[i] = src[i];
  rec[REC - 1] = 0.f;            // zero slot for padded/invalid rows
  wait_ds();                     // intra-wave: all lanes' stores visible
}

// A[m][k] = staged[pos = kt*4+k][aoff(m)], B[k][n] = staged[pos][boff(n)]
// 32-bit A layout: lanes 0-15 K=0,1 (VGPR0,1); lanes 16-31 K=2,3. B mirrored with N=lane%16.
__device__ __forceinline__ v8f gram_tiles(const float* slab, int hi2, int aoff, int boff, v8f acc) {
#pragma unroll
  for (int kt = 0; kt < 8; ++kt) {
    int p0 = kt * 4 + hi2;
    v2f a, bm;
    a.x  = slab[(p0 + 0) * REC + aoff];
    a.y  = slab[(p0 + 1) * REC + aoff];
    bm.x = slab[(p0 + 0) * REC + boff];
    bm.y = slab[(p0 + 1) * REC + boff];
    acc = __builtin_amdgcn_wmma_f32_16x16x4_f32(false, a, false, bm, (short)0, acc,
                                                false, false);
  }
  wait_ds();
  return acc;
}

extern "C" __global__ void __launch_bounds__(256)
k_zero(float* ws) {
  int i = blockIdx.x * 256 + threadIdx.x;
  if (i < WS_FLOATS) ws[i] = 0.f;
}

// Grid: 8 batches x 32 subtrees x 13 position-chunks; one position per thread (padded, zeros inert).
extern "C" __global__ void __launch_bounds__(256)
k_main(const float* __restrict__ x,
       const float* __restrict__ P1a, const float* __restrict__ P1b,
       const float* __restrict__ P2a, const float* __restrict__ P2b,
       const float* __restrict__ P3a, const float* __restrict__ P3b,
       float* __restrict__ ws)
{
  __shared__ float sP[344];                 // m-chain sketch matrices
  __shared__ float sSlab[8 * 32 * REC];     // per-wave transpose slabs

  const int tid = threadIdx.x;
  for (int i = tid; i < 344; i += 256) {
    float v;
    if (i < 12)       v = P1a[i];
    else if (i < 24)  v = P1b[i - 12];
    else if (i < 56)  v = P2a[i - 24];
    else if (i < 88)  v = P2b[i - 56];
    else if (i < 216) v = P3a[i - 88];
    else              v = P3b[i - 216];
    sP[i] = v;
  }
  __syncthreads();

  const int blk = blockIdx.x;
  const int b = blk / (32 * 13);
  const int rem = blk % (32 * 13);
  const int t = rem / 13;
  const int chunk = rem % 13;
  const int pos = chunk * 256 + tid;
  const bool valid = pos < 3136;

  const int lane = tid & 31;
  const int wv = tid >> 5;
  float* slab = sSlab + wv * 32 * REC;
  const int hi2 = (lane >> 4) << 1;     // K offset for upper lane half
  const int n = lane & 15;              // A-row index for loads == C-column for epilogue
  const int un = n / 3, cn = n - 3 * un;

  // Per-level record offsets (block-diagonal packing of sub-pair Grams into 16x16)
  const int a1a = (n < 15) ? (2 * un + cn)         : (REC - 1);   // v1 pairs 0..4
  const int b1a = (n < 15) ? (2 * un + 1 + cn)     : (REC - 1);
  const int a1b = (n < 9)  ? (2 * (un + 5) + cn)   : (REC - 1);   // v1 pairs 5..7
  const int b1b = (n < 9)  ? (2 * (un + 5) + 1 + cn) : (REC - 1);
  const int a2 = (n >> 2) * 8 + (n & 3),  b2o = a2 + 4;           // v2: 4 Grams of 4x4
  const int a3 = (n >> 3) * 16 + (n & 7), b3o = a3 + 8;           // v3: 2 Grams of 8x8
  const int a4 = n,                       b4o = n + 16;           // v4: one 16x16

  // x window: 18 channels starting at 16t (only channels 0..513 of x are ever used)
  float xw[18];
  const size_t xbase = (size_t)(b * 1536 + 16 * t) * 3136 + (size_t)pos;
#pragma unroll
  for (int c = 0; c < 18; ++c)
    xw[c] = valid ? x[xbase + (size_t)c * 3136] : 0.f;

  v8f c1a = {0,0,0,0,0,0,0,0}, c1b = {0,0,0,0,0,0,0,0};
  v8f c2v = {0,0,0,0,0,0,0,0}, c3v = {0,0,0,0,0,0,0,0}, c4v = {0,0,0,0,0,0,0,0};

  // ---- level 1: v1 Grams from raw x; m1 via cbp+ssn ----
  stage<18>(slab, lane, xw);
  c1a = gram_tiles(slab, hi2, a1a, b1a, c1a);
  c1b = gram_tiles(slab, hi2, a1b, b1b, c1b);

  float m1[32];
#pragma unroll
  for (int p = 0; p < 8; ++p) {
    float s1[4], s2[4];
    proj<3, 4>(sP + 0,  xw + 2 * p,     s1);
    proj<3, 4>(sP + 12, xw + 2 * p + 1, s2);
    cconv<4>(s1, s2, m1 + 4 * p);
    ssn_norm<4>(m1 + 4 * p);
  }

  // ---- level 2 ----
  stage<32>(slab, lane, m1);
  c2v = gram_tiles(slab, hi2, a2, b2o, c2v);

  float m2[32];
#pragma unroll
  for (int q = 0; q < 4; ++q) {
    float s1[8], s2[8];
    proj<4, 8>(sP + 24, m1 + 8 * q,     s1);
    proj<4, 8>(sP + 56, m1 + 8 * q + 4, s2);
    cconv<8>(s1, s2, m2 + 8 * q);
    ssn_norm<8>(m2 + 8 * q);
  }

  // ---- level 3 ----
  stage<32>(slab, lane, m2);
  c3v = gram_tiles(slab, hi2, a3, b3o, c3v);

  float m3[32];
#pragma unroll
  for (int r = 0; r < 2; ++r) {
    float s1[16], s2[16];
    proj<8, 16>(sP + 88,  m2 + 16 * r,     s1);
    proj<8, 16>(sP + 216, m2 + 16 * r + 8, s2);
    cconv<16>(s1, s2, m3 + 16 * r);
    ssn_norm<16>(m3 + 16 * r);
  }

  // ---- level 4 ----
  stage<32>(slab, lane, m3);
  c4v = gram_tiles(slab, hi2, a4, b4o, c4v);

  // ---- epilogue: scatter valid Gram entries (C layout: VGPR j -> M=j or j+8) ----
  float* G1 = ws + G1_OFF;
  float* G2 = ws + G2_OFF;
  float* G3 = ws + G3_OFF;
  float* G4 = ws + G4_OFF;
  const int mh = (lane >> 4) * 8;
#pragma unroll
  for (int j = 0; j < 8; ++j) {
    const int mm = j + mh;
    atomAddF(&G4[(size_t)(b * 32 + t) * 256 + mm * 16 + n], c4v[j]);
    if ((mm >> 3) == (n >> 3)) {
      int r = 2 * t + (mm >> 3);
      atomAddF(&G3[(size_t)(b * 64 + r) * 64 + (mm & 7) * 8 + (n & 7)], c3v[j]);
    }
    if ((mm >> 2) == (n >> 2)) {
      int q = 4 * t + (mm >> 2);
      atomAddF(&G2[(size_t)(b * 128 + q) * 16 + (mm & 3) * 4 + (n & 3)], c2v[j]);
    }
    int umm = mm / 3, cmm = mm - 3 * umm;
    if (mm < 15 && n < 15 && umm == un)
      atomAddF(&G1[(size_t)(b * 256 + 8 * t + umm) * 9 + cmm * 3 + cn], c1a[j]);
    if (mm < 9 && n < 9 && umm == un)
      atomAddF(&G1[(size_t)(b * 256 + 8 * t + 5 + umm) * 9 + cmm * 3 + cn], c1b[j]);
  }
}

// ---- kernel 2: count-sketch contraction v_pre[d] = sum T[d,c1,c2]*G[c1,c2], then ssn ----
template<int C, int D>
__device__ void finish_one(const float* __restrict__ G,
                           const float* __restrict__ Pa, const float* __restrict__ Pb,
                           float* __restrict__ op) {
  int ha[C]; float sa[C]; int hb[C]; float sb[C];
  for (int c = 0; c < C; ++c) {
    int h = 0; float s = 0.f;
    for (int d = 0; d < D; ++d) { float v = Pa[c * D + d]; if (v != 0.f) { h = d; s = v; } }
    ha[c] = h; sa[c] = s;
    h = 0; s = 0.f;
    for (int d = 0; d < D; ++d) { float v = Pb[c * D + d]; if (v != 0.f) { h = d; s = v; } }
    hb[c] = h; sb[c] = s;
  }
  float acc[D];
  for (int d = 0; d < D; ++d) acc[d] = 0.f;
  for (int c1 = 0; c1 < C; ++c1)
    for (int c2 = 0; c2 < C; ++c2) {
      int dd = ha[c1] + hb[c2]; if (dd >= D) dd -= D;
      acc[dd] += sa[c1] * sb[c2] * G[c1 * C + c2];
    }
  float n2 = 0.f; float y[D];
  for (int d = 0; d < D; ++d) {
    float tv = acc[d];
    float s = sqrtf(fabsf(tv) + EPS_SQ);
    float yy = tv > 0.f ? s : (tv < 0.f ? -s : 0.f);
    y[d] = yy; n2 += yy * yy;
  }
  float inv = 1.f / fmaxf(sqrtf(n2), 1e-12f);
  for (int d = 0; d < D; ++d) op[d] = y[d] * inv;
}

extern "C" __global__ void __launch_bounds__(512)
k_finish(const float* __restrict__ ws,
         const float* __restrict__ P1va, const float* __restrict__ P1vb,
         const float* __restrict__ P2va, const float* __restrict__ P2vb,
         const float* __restrict__ P3va, const float* __restrict__ P3vb,
         const float* __restrict__ P4va, const float* __restrict__ P4vb,
         float* __restrict__ out)
{
  const int b = blockIdx.x;
  const int j = threadIdx.x;
  const float* G1 = ws + G1_OFF;
  const float* G2 = ws + G2_OFF;
  const float* G3 = ws + G3_OFF;
  const float* G4 = ws + G4_OFF;
  float* ob = out + (size_t)b * 4096;
  if (j < 256) {
    finish_one<3, 4>(G1 + (size_t)(b * 256 + j) * 9, P1va, P1vb, ob + j * 4);
  } else if (j < 384) {
    int q = j - 256;
    finish_one<4, 8>(G2 + (size_t)(b * 128 + q) * 16, P2va, P2vb, ob + 1024 + q * 8);
  } else if (j < 448) {
    int r = j - 384;
    finish_one<8, 16>(G3 + (size_t)(b * 64 + r) * 64, P3va, P3vb, ob + 2048 + r * 16);
  } else if (j < 480) {
    int tt = j - 448;
    finish_one<16, 32>(G4 + (size_t)(b * 32 + tt) * 256, P4va, P4vb, ob + 3072 + tt * 32);
  }
}

extern "C" void kernel_launch(void* const* d_in, const int* in_sizes, int n_in,
                              void* d_out, int out_size, void* d_ws, size_t ws_size,
                              hipStream_t stream) {
  (void)in_sizes; (void)n_in; (void)out_size; (void)ws_size;
  const float* x    = (const float*)d_in[0];
  const float* P1a  = (const float*)d_in[1];
  const float* P1b  = (const float*)d_in[2];
  const float* P1va = (const float*)d_in[3];
  const float* P1vb = (const float*)d_in[4];
  const float* P2a  = (const float*)d_in[5];
  const float* P2b  = (const float*)d_in[6];
  const float* P2va = (const float*)d_in[7];
  const float* P2vb = (const float*)d_in[8];
  const float* P3a  = (const float*)d_in[9];
  const float* P3b  = (const float*)d_in[10];
  const float* P3va = (const float*)d_in[11];
  const float* P3vb = (const float*)d_in[12];
  const float* P4va = (const float*)d_in[13];
  const float* P4vb = (const float*)d_in[14];
  float* ws  = (float*)d_ws;
  float* out = (float*)d_out;

  k_zero<<<dim3((WS_FLOATS + 255) / 256), dim3(256), 0, stream>>>(ws);
  k_main<<<dim3(8 * 32 * 13), dim3(256), 0, stream>>>(x, P1a, P1b, P2a, P2b, P3a, P3b, ws);
  k_finish<<<dim3(8), dim3(512), 0, stream>>>(ws, P1va, P1vb, P2va, P2vb,
                                              P3va, P3vb, P4va, P4vb, out);
}